// GrapherDGL_3135326126137
// MI455X (gfx1250) — compile-verified
//
#include <hip/hip_runtime.h>

typedef __attribute__((ext_vector_type(2))) float v2f;
typedef __attribute__((ext_vector_type(4))) float v4f;
typedef __attribute__((ext_vector_type(8))) float v8f;

#define CCH 128      // channels C
#define LDSP 130     // padded LDS row stride (floats) -> conflict-free

// ---------------------------------------------------------------------------
// Kernel 0: fuse weights.  W is [C, 2C] row-major.  Build Wc [2C, C]:
//   Wc[c]     = W[c, 0:C] - W[c, C:2C]   (multiplies x_dst)
//   Wc[C + c] = W[c, C:2C]               (multiplies x_src)
// ---------------------------------------------------------------------------
__global__ __launch_bounds__(256) void prep_w(const float* __restrict__ W,
                                              float* __restrict__ Wc) {
  int i = blockIdx.x * 256 + threadIdx.x;          // 0 .. C*C-1
  int c = i >> 7, k = i & (CCH - 1);
  float w1 = W[c * (2 * CCH) + k];
  float w2 = W[c * (2 * CCH) + CCH + k];
  Wc[c * CCH + k] = w1 - w2;
  Wc[(CCH + c) * CCH + k] = w2;
}

// ---------------------------------------------------------------------------
// Kernel 1: zero the per-node degree histogram
// ---------------------------------------------------------------------------
__global__ __launch_bounds__(256) void zero_count(int* __restrict__ cnt,
                                                  int N) {
  int i = blockIdx.x * 256 + threadIdx.x;
  if (i < N) cnt[i] = 0;
}

// ---------------------------------------------------------------------------
// Kernel 2: degree histogram over edge destinations (0.8M u32 atomics)
// ---------------------------------------------------------------------------
__global__ __launch_bounds__(256) void count_deg(const int* __restrict__ ei,
                                                 int* __restrict__ cnt,
                                                 int E) {
  int e = blockIdx.x * 256 + threadIdx.x;
  if (e < E) atomicAdd(&cnt[ei[E + e]], 1);
}

// ---------------------------------------------------------------------------
// Kernel 3: single-block exclusive prefix scan (shuffle-based wave scan,
// 4 barriers per 1024-element chunk).  Writes offs[] and a scratch cursor[].
// ---------------------------------------------------------------------------
__device__ __forceinline__ int wave_incl_scan(int v, int lane) {
#pragma unroll
  for (int d = 1; d < 32; d <<= 1) {
    int t = __shfl_up(v, d, 32);
    if (lane >= d) v += t;
  }
  return v;
}

__global__ __launch_bounds__(1024) void scan_offsets(
    const int* __restrict__ cnt, int* __restrict__ offs,
    int* __restrict__ cursor, int N) {
  __shared__ int wsum[32];
  __shared__ int carry_s;
  const int tid = threadIdx.x;
  const int lane = tid & 31;
  const int wid = tid >> 5;
  if (tid == 0) carry_s = 0;
  __syncthreads();
  for (int base = 0; base < N; base += 1024) {
    int i = base + tid;
    int v = (i < N) ? cnt[i] : 0;
    int incl = wave_incl_scan(v, lane);
    if (lane == 31) wsum[wid] = incl;
    __syncthreads();
    int carry = carry_s;
    if (wid == 0) wsum[lane] = wave_incl_scan(wsum[lane], lane);
    __syncthreads();
    int wave_excl = (wid == 0) ? 0 : wsum[wid - 1];
    int excl = carry + wave_excl + incl - v;
    if (i < N) {
      offs[i] = excl;
      cursor[i] = excl;
    }
    int total = wsum[31];
    __syncthreads();
    if (tid == 0) carry_s = carry + total;
    __syncthreads();
  }
}

// ---------------------------------------------------------------------------
// Kernel 4: scatter sources into CSR adjacency list (0.8M atomic-add-return)
// ---------------------------------------------------------------------------
__global__ __launch_bounds__(256) void scatter_edges(
    const int* __restrict__ ei, int* __restrict__ cursor,
    int* __restrict__ list, int E) {
  int e = blockIdx.x * 256 + threadIdx.x;
  if (e < E) {
    int d = ei[E + e];
    int pos = atomicAdd(&cursor[d], 1);
    list[pos] = ei[e];
  }
}

// ---------------------------------------------------------------------------
// Kernel 5: A = x @ (W1-W2)^T, B = x @ W2^T via V_WMMA_F32_16X16X4_F32.
// One block = one 16-row tile of x (staged in LDS), 8 waves cover the 16
// column tiles of the fused [2C] output (t<8 -> A, t>=8 -> B).
// Fragment layouts per CDNA5 ISA 7.12.2.
// ---------------------------------------------------------------------------
__global__ __launch_bounds__(256) void fused_gemm(const float* __restrict__ x,
                                                  const float* __restrict__ Wc,
                                                  float* __restrict__ A,
                                                  float* __restrict__ B,
                                                  int N) {
  __shared__ float xs[16 * LDSP];
  const int tid = threadIdx.x;
  const int row0 = blockIdx.x * 16;

  for (int i = tid; i < 16 * CCH; i += 256) {
    int r = i >> 7, col = i & (CCH - 1);
    int gr = row0 + r;
    if (gr >= N) gr = N - 1;
    xs[r * LDSP + col] = x[(size_t)gr * CCH + col];
  }
  __syncthreads();

  const int wid = tid >> 5;
  const int lane = tid & 31;
  const int half = lane >> 4;
  const int n = lane & 15;
  const bool full = (row0 + 16 <= N);   // block-uniform fast path

  for (int t = wid; t < 16; t += 8) {
    const float* wcol = Wc + (size_t)(t * 16 + n) * CCH;
    v8f acc = {};
#pragma unroll 8
    for (int k = 0; k < CCH; k += 4) {
      int kk = k + (half << 1);
      v2f a = *(const v2f*)(xs + n * LDSP + kk);
      v2f bf = *(const v2f*)(wcol + kk);
      acc = __builtin_amdgcn_wmma_f32_16x16x4_f32(
          false, a, false, bf, (short)0, acc, false, false);
    }
    int cg = t * 16 + n;
    float* dst = (cg < CCH) ? (A + cg) : (B + (cg - CCH));
    int mb = half * 8;
    if (full) {
#pragma unroll
      for (int v = 0; v < 8; ++v)
        dst[(size_t)(row0 + mb + v) * CCH] = acc[v];
    } else {
#pragma unroll
      for (int v = 0; v < 8; ++v) {
        int gr = row0 + mb + v;
        if (gr < N) dst[(size_t)gr * CCH] = acc[v];
      }
    }
  }
}

// ---------------------------------------------------------------------------
// Kernel 6: fused gather + segment-max + bias + ReLU.  One wave per node,
// lane holds 4 channels (b128 loads), running max over the node's CSR list.
// deg==0 nodes -> 0 (matches reference's isfinite() guard).  No atomics.
// ---------------------------------------------------------------------------
__device__ __forceinline__ v4f max4(v4f a, v4f b) {
  v4f r;
  r.x = fmaxf(a.x, b.x);
  r.y = fmaxf(a.y, b.y);
  r.z = fmaxf(a.z, b.z);
  r.w = fmaxf(a.w, b.w);
  return r;
}

__global__ __launch_bounds__(256) void gather_max_relu(
    const float* __restrict__ A, const float* __restrict__ B,
    const float* __restrict__ bias, const int* __restrict__ offs,
    const int* __restrict__ cnt, const int* __restrict__ list,
    float* __restrict__ out, int N) {
  int node = (int)((blockIdx.x * 256 + threadIdx.x) >> 5);
  if (node >= N) return;
  int lane = threadIdx.x & 31;
  int c4 = lane * 4;

  int beg = __builtin_amdgcn_readfirstlane(offs[node]);
  int deg = __builtin_amdgcn_readfirstlane(cnt[node]);

  float ninf = -__builtin_inff();
  v4f m = {ninf, ninf, ninf, ninf};
  int j = 0;
  for (; j + 2 <= deg; j += 2) {          // unroll-by-2 to hide L2 latency
    int s0 = __builtin_amdgcn_readfirstlane(list[beg + j]);
    int s1 = __builtin_amdgcn_readfirstlane(list[beg + j + 1]);
    v4f v0 = *(const v4f*)(B + (size_t)s0 * CCH + c4);
    v4f v1 = *(const v4f*)(B + (size_t)s1 * CCH + c4);
    m = max4(m, max4(v0, v1));
  }
  if (j < deg) {
    int s0 = __builtin_amdgcn_readfirstlane(list[beg + j]);
    m = max4(m, *(const v4f*)(B + (size_t)s0 * CCH + c4));
  }

  v4f a = *(const v4f*)(A + (size_t)node * CCH + c4);
  v4f bb = *(const v4f*)(bias + c4);
  v4f r;
  if (deg > 0) {
    r.x = fmaxf(a.x + bb.x + m.x, 0.0f);
    r.y = fmaxf(a.y + bb.y + m.y, 0.0f);
    r.z = fmaxf(a.z + bb.z + m.z, 0.0f);
    r.w = fmaxf(a.w + bb.w + m.w, 0.0f);
  } else {
    r.x = r.y = r.z = r.w = 0.0f;
  }
  *(v4f*)(out + (size_t)node * CCH + c4) = r;
}

// ---------------------------------------------------------------------------
extern "C" void kernel_launch(void* const* d_in, const int* in_sizes, int n_in,
                              void* d_out, int out_size, void* d_ws,
                              size_t ws_size, hipStream_t stream) {
  const float* x = (const float*)d_in[0];   // [N, C]
  const float* W = (const float*)d_in[1];   // [C, 2C]
  const float* b = (const float*)d_in[2];   // [C]
  const int* ei = (const int*)d_in[3];      // [2, E]
  float* out = (float*)d_out;               // [N, C]

  const int N = in_sizes[0] / CCH;
  const int E = in_sizes[3] / 2;
  const size_t nc = (size_t)N * CCH;

  // workspace layout: Wc[2C*C] f32 | A[N*C] f32 | B[N*C] f32 |
  //                   cnt[N] i32 | offs[N] i32 | cursor[N] i32 | list[E] i32
  float* ws = (float*)d_ws;
  float* Wc = ws;
  float* A = Wc + (size_t)2 * CCH * CCH;
  float* Bm = A + nc;
  int* cnt = (int*)(Bm + nc);
  int* offs = cnt + N;
  int* cursor = offs + N;
  int* list = cursor + N;

  const int gE = (E + 255) / 256;
  const int gN = (N + 255) / 256;

  prep_w<<<(CCH * CCH) / 256, 256, 0, stream>>>(W, Wc);
  zero_count<<<gN, 256, 0, stream>>>(cnt, N);
  count_deg<<<gE, 256, 0, stream>>>(ei, cnt, E);
  scan_offsets<<<1, 1024, 0, stream>>>(cnt, offs, cursor, N);
  scatter_edges<<<gE, 256, 0, stream>>>(ei, cursor, list, E);
  fused_gemm<<<(N + 15) / 16, 256, 0, stream>>>(x, Wc, A, Bm, N);
  gather_max_relu<<<(N * 32 + 255) / 256, 256, 0, stream>>>(
      A, Bm, b, offs, cnt, list, out, N);
}